// MLPNet_43198781063214
// MI455X (gfx1250) — compile-verified
//
#include <hip/hip_runtime.h>
#include <math.h>

#define N_NODES 100000
#define E_EDGES 3200000
#define CY 128
#define HID 10
#define HC 20            // heads * out-channels
#define MID 512

// ws layout (float offsets)
#define H_OFF   0                              // N*16
#define XL_OFF  (N_NODES*16)                   // N*20
#define XR_OFF  (XL_OFF + N_NODES*20)          // N*20
#define EX_OFF  (XR_OFF + N_NODES*20)          // E*2
#define DEN_OFF (EX_OFF + E_EDGES*2)           // N*2
#define OG_OFF  (DEN_OFF + N_NODES*2)          // N*20

#define LDS_STRIDE 516   // 516 % 64 == 4 -> conflict-free column reads

typedef __attribute__((ext_vector_type(2))) float v2f;
typedef __attribute__((ext_vector_type(8))) float v8f;

__device__ __forceinline__ v8f wmma4(v2f a, v2f b, v8f c) {
    // D = A(16x4,f32) x B(4x16,f32) + C(16x16,f32)
    return __builtin_amdgcn_wmma_f32_16x16x4_f32(
        false, a, false, b, (short)0, c, false, false);
}

// ---------------------------------------------------------------------------
// Stage 1: h = relu(bn(x_masked @ W1 + b1)) @ W2 + b2      h:[N,16] (10 valid)
// One block = 16 nodes, 8 waves.
// ---------------------------------------------------------------------------
__global__ __launch_bounds__(256) void k_mlp_in(
    const float* __restrict__ signal, const unsigned char* __restrict__ mask,
    const float* __restrict__ W1, const float* __restrict__ b1,
    const float* __restrict__ gamma, const float* __restrict__ beta,
    const float* __restrict__ W2, const float* __restrict__ b2,
    float* __restrict__ h)
{
    __shared__ float ldsT[16 * LDS_STRIDE];   // 16x512 intermediate (padded)
    __shared__ float ldsR[8 * 256];           // per-wave partial 16x16 tiles

    const int tid   = threadIdx.x;
    const int lane  = tid & 31;
    const int wave  = tid >> 5;
    const int r16   = lane & 15;
    const int khalf = lane >> 4;          // 0: K even pair, 1: K odd pair
    const int row0  = blockIdx.x * 16;
    const int row   = row0 + r16;

    const float RS = 0.9999950000375f;    // 1/sqrt(1 + 1e-5)
    const float msc = mask[row] ? 0.0f : 1.0f;

    // A fragments for the whole K=128 (reused across 4 column tiles)
    v2f af[32];
    const float* xr = signal + (size_t)row * CY;
    #pragma unroll
    for (int s = 0; s < 32; ++s) {
        const int k = s * 4 + khalf * 2;
        af[s][0] = xr[k]     * msc;
        af[s][1] = xr[k + 1] * msc;
    }

    // ---- GEMM1: 16x128 @ 128x512, fused bias+BN+ReLU -> LDS ----
    for (int t = 0; t < 4; ++t) {
        const int n0  = (wave * 4 + t) * 16;
        const int col = n0 + r16;
        v8f acc = {};
        #pragma unroll
        for (int s = 0; s < 32; ++s) {
            const int kb = s * 4 + khalf * 2;
            v2f bf;
            bf[0] = W1[(size_t)kb * MID + col];
            bf[1] = W1[(size_t)(kb + 1) * MID + col];
            acc = wmma4(af[s], bf, acc);
        }
        const float sc = gamma[col] * RS;
        const float bt = beta[col];
        const float bb = b1[col];
        #pragma unroll
        for (int v = 0; v < 8; ++v) {
            const int rm = v + khalf * 8;
            float val = (acc[v] + bb) * sc + bt;
            val = val > 0.0f ? val : 0.0f;
            ldsT[rm * LDS_STRIDE + col] = val;
        }
    }
    __syncthreads();

    // ---- GEMM2: 16x512 @ 512x10 (cols padded to 16), K split across waves --
    {
        const int kbase = wave * 64;
        v8f acc = {};
        #pragma unroll
        for (int s = 0; s < 16; ++s) {
            const int k = kbase + s * 4 + khalf * 2;
            v2f a2, b2f;
            a2[0] = ldsT[r16 * LDS_STRIDE + k];
            a2[1] = ldsT[r16 * LDS_STRIDE + k + 1];
            b2f[0] = (r16 < HID) ? W2[k * HID + r16]       : 0.0f;
            b2f[1] = (r16 < HID) ? W2[(k + 1) * HID + r16] : 0.0f;
            acc = wmma4(a2, b2f, acc);
        }
        #pragma unroll
        for (int v = 0; v < 8; ++v) {
            const int rm = v + khalf * 8;
            ldsR[wave * 256 + rm * 16 + r16] = acc[v];
        }
    }
    __syncthreads();

    if (tid < 16 * HID) {
        const int rr = tid / HID, cc = tid - rr * HID;
        float s = b2[cc];
        #pragma unroll
        for (int w = 0; w < 8; ++w) s += ldsR[w * 256 + rr * 16 + cc];
        h[(size_t)(row0 + rr) * 16 + cc] = s;
    }
}

// ---------------------------------------------------------------------------
// Stage 2: xl = h@Wl + bl, xr = h@Wr + br    (K=10, tiny -> VALU)
// ---------------------------------------------------------------------------
__global__ __launch_bounds__(256) void k_gat_lin(
    const float* __restrict__ h,
    const float* __restrict__ Wl, const float* __restrict__ bl,
    const float* __restrict__ Wr, const float* __restrict__ br,
    float* __restrict__ xl, float* __restrict__ xr)
{
    const int idx = blockIdx.x * blockDim.x + threadIdx.x;
    if (idx >= N_NODES * HC) return;
    const int n = idx / HC, j = idx - n * HC;
    const float* hr = h + (size_t)n * 16;
    float sl = bl[j], sr = br[j];
    #pragma unroll
    for (int k = 0; k < HID; ++k) {
        const float hv = hr[k];
        sl += hv * Wl[k * HC + j];
        sr += hv * Wr[k * HC + j];
    }
    xl[(size_t)n * HC + j] = sl;
    xr[(size_t)n * HC + j] = sr;
}

// ---------------------------------------------------------------------------
// Stage 3a: per-edge attention logits -> exp, accumulate softmax denominator
// ---------------------------------------------------------------------------
__global__ __launch_bounds__(256) void k_edge_alpha(
    const int* __restrict__ edge_index, const float* __restrict__ edge_weight,
    const float* __restrict__ xl, const float* __restrict__ xr,
    const float* __restrict__ We, const float* __restrict__ att,
    float* __restrict__ ex, float* __restrict__ den)
{
    const int e = blockIdx.x * blockDim.x + threadIdx.x;
    if (e >= E_EDGES) return;
    const int src = edge_index[e];
    const int dst = edge_index[E_EDGES + e];
    const float w = edge_weight[e];
    const float* xls = xl + (size_t)src * HC;
    const float* xrd = xr + (size_t)dst * HC;
    #pragma unroll
    for (int hh = 0; hh < 2; ++hh) {
        float alpha = 0.0f;
        #pragma unroll
        for (int c = 0; c < HID; ++c) {
            const int j = hh * HID + c;
            float v = xls[j] + xrd[j] + w * We[j];
            v = v > 0.0f ? v : 0.2f * v;          // leaky_relu(0.2)
            alpha += v * att[j];
        }
        const float ev = expf(alpha);
        ex[(size_t)e * 2 + hh] = ev;
        atomicAdd(&den[(size_t)dst * 2 + hh], ev);
    }
}

// ---------------------------------------------------------------------------
// Stage 3b: weighted scatter-add: og[dst] += xl[src] * softmax_weight
// ---------------------------------------------------------------------------
__global__ __launch_bounds__(256) void k_edge_aggr(
    const int* __restrict__ edge_index,
    const float* __restrict__ xl, const float* __restrict__ ex,
    const float* __restrict__ den, float* __restrict__ og)
{
    const int e = blockIdx.x * blockDim.x + threadIdx.x;
    if (e >= E_EDGES) return;
    const int src = edge_index[e];
    const int dst = edge_index[E_EDGES + e];
    const float* xls = xl + (size_t)src * HC;
    float* ogd = og + (size_t)dst * HC;
    #pragma unroll
    for (int hh = 0; hh < 2; ++hh) {
        const float a = ex[(size_t)e * 2 + hh] /
                        (den[(size_t)dst * 2 + hh] + 1e-16f);
        #pragma unroll
        for (int c = 0; c < HID; ++c) {
            const int j = hh * HID + c;
            atomicAdd(&ogd[j], xls[j] * a);
        }
    }
}

// ---------------------------------------------------------------------------
// Stage 4: out = ((og + gat_bias) @ W3 + b3) @ W4 + b4     out:[N,128]
// One block = 16 nodes, 8 waves.
// ---------------------------------------------------------------------------
__global__ __launch_bounds__(256) void k_mlp_out(
    const float* __restrict__ og, const float* __restrict__ gat_bias,
    const float* __restrict__ W3, const float* __restrict__ b3,
    const float* __restrict__ W4, const float* __restrict__ b4,
    float* __restrict__ out)
{
    __shared__ float ldsU[16 * LDS_STRIDE];   // 16x512 intermediate

    const int tid   = threadIdx.x;
    const int lane  = tid & 31;
    const int wave  = tid >> 5;
    const int r16   = lane & 15;
    const int khalf = lane >> 4;
    const int row0  = blockIdx.x * 16;
    const int row   = row0 + r16;

    // A fragments: K = 20 = 5 steps of 4
    v2f af[5];
    const float* ar = og + (size_t)row * HC;
    #pragma unroll
    for (int s = 0; s < 5; ++s) {
        const int k = s * 4 + khalf * 2;
        af[s][0] = ar[k]     + gat_bias[k];
        af[s][1] = ar[k + 1] + gat_bias[k + 1];
    }

    // ---- GEMM1: 16x20 @ 20x512 + b3 -> LDS ----
    for (int t = 0; t < 4; ++t) {
        const int n0  = (wave * 4 + t) * 16;
        const int col = n0 + r16;
        v8f acc = {};
        #pragma unroll
        for (int s = 0; s < 5; ++s) {
            const int k = s * 4 + khalf * 2;
            v2f bf;
            bf[0] = W3[(size_t)k * MID + col];
            bf[1] = W3[(size_t)(k + 1) * MID + col];
            acc = wmma4(af[s], bf, acc);
        }
        const float bb = b3[col];
        #pragma unroll
        for (int v = 0; v < 8; ++v)
            ldsU[(v + khalf * 8) * LDS_STRIDE + col] = acc[v] + bb;
    }
    __syncthreads();

    // ---- GEMM2: 16x512 @ 512x128 + b4 -> out; one 16-col tile per wave ----
    {
        const int col = wave * 16 + r16;
        v8f acc = {};
        for (int s = 0; s < 128; ++s) {
            const int k = s * 4 + khalf * 2;
            if ((s & 15) == 0 && k + 64 < MID)
                __builtin_prefetch(&W4[(size_t)(k + 64) * CY + col], 0, 1);
            v2f a2, bf;
            a2[0] = ldsU[r16 * LDS_STRIDE + k];
            a2[1] = ldsU[r16 * LDS_STRIDE + k + 1];
            bf[0] = W4[(size_t)k * CY + col];
            bf[1] = W4[(size_t)(k + 1) * CY + col];
            acc = wmma4(a2, bf, acc);
        }
        const float bb = b4[col];
        #pragma unroll
        for (int v = 0; v < 8; ++v)
            out[(size_t)(row0 + v + khalf * 8) * CY + col] = acc[v] + bb;
    }
}

// ---------------------------------------------------------------------------
extern "C" void kernel_launch(void* const* d_in, const int* in_sizes, int n_in,
                              void* d_out, int out_size, void* d_ws, size_t ws_size,
                              hipStream_t stream) {
    const float*         signal      = (const float*)d_in[0];
    const int*           edge_index  = (const int*)d_in[1];
    const float*         edge_weight = (const float*)d_in[2];
    const unsigned char* mask        = (const unsigned char*)d_in[3];
    const float* W1    = (const float*)d_in[4];
    const float* b1    = (const float*)d_in[5];
    const float* gamma = (const float*)d_in[6];
    const float* beta  = (const float*)d_in[7];
    const float* W2    = (const float*)d_in[8];
    const float* b2    = (const float*)d_in[9];
    const float* Wl    = (const float*)d_in[10];
    const float* bl    = (const float*)d_in[11];
    const float* Wr    = (const float*)d_in[12];
    const float* br    = (const float*)d_in[13];
    const float* We    = (const float*)d_in[14];
    const float* att   = (const float*)d_in[15];
    const float* gbias = (const float*)d_in[16];
    const float* W3    = (const float*)d_in[17];
    const float* b3    = (const float*)d_in[18];
    const float* W4    = (const float*)d_in[19];
    const float* b4    = (const float*)d_in[20];

    float* ws  = (float*)d_ws;
    float* h   = ws + H_OFF;
    float* xl  = ws + XL_OFF;
    float* xr  = ws + XR_OFF;
    float* ex  = ws + EX_OFF;
    float* den = ws + DEN_OFF;
    float* og  = ws + OG_OFF;

    // zero softmax denominators + aggregation buffer (contiguous region)
    hipMemsetAsync(den, 0, (size_t)(N_NODES * 2 + N_NODES * HC) * sizeof(float),
                   stream);

    k_mlp_in<<<N_NODES / 16, 256, 0, stream>>>(signal, mask, W1, b1, gamma, beta,
                                               W2, b2, h);
    k_gat_lin<<<(N_NODES * HC + 255) / 256, 256, 0, stream>>>(h, Wl, bl, Wr, br,
                                                              xl, xr);
    k_edge_alpha<<<(E_EDGES + 255) / 256, 256, 0, stream>>>(edge_index,
                                                            edge_weight, xl, xr,
                                                            We, att, ex, den);
    k_edge_aggr<<<(E_EDGES + 255) / 256, 256, 0, stream>>>(edge_index, xl, ex,
                                                           den, og);
    k_mlp_out<<<N_NODES / 16, 256, 0, stream>>>(og, gbias, W3, b3, W4, b4,
                                                (float*)d_out);
}